// SimpleNet_79302276153656
// MI455X (gfx1250) — compile-verified
//
#include <hip/hip_runtime.h>
#include <hip/hip_bf16.h>

#define N_NODES 100000
#define N_EDGES 1600000
#define N_GRAPHS 64
#define IN_DIM 128
#define HID 64

typedef __attribute__((ext_vector_type(2))) float v2f;
typedef __attribute__((ext_vector_type(8))) float v8f;
typedef __attribute__((ext_vector_type(4))) int v4i;

#define GLOBAL_AS __attribute__((address_space(1)))
#define LDS_AS __attribute__((address_space(3)))

#if __has_builtin(__builtin_amdgcn_global_load_async_to_lds_b128) && \
    __has_builtin(__builtin_amdgcn_s_wait_asynccnt)
#define USE_ASYNC_LDS 1
#else
#define USE_ASYNC_LDS 0
#endif

// D = A(16x4 f32) * B(4x16 f32) + C(16x16 f32), exact fp32 on the matrix pipe.
__device__ __forceinline__ v8f wmma_f32(v2f a, v2f b, v8f c) {
  return __builtin_amdgcn_wmma_f32_16x16x4_f32(
      /*neg_a=*/false, a, /*neg_b=*/false, b,
      /*c_mod=*/(short)0, c, /*reuse_a=*/false, /*reuse_b=*/false);
}

// Async global->LDS 128b copy (ASYNCcnt path); falls back to VALU staging.
__device__ __forceinline__ void stage_b128(const float4* g, float* lds) {
#if USE_ASYNC_LDS
  __builtin_amdgcn_global_load_async_to_lds_b128(
      (GLOBAL_AS v4i*)(v4i*)g, (LDS_AS v4i*)(v4i*)lds, 0, 0);
#else
  *(float4*)lds = *g;
#endif
}

__device__ __forceinline__ void stage_wait() {
#if USE_ASYNC_LDS
  __builtin_amdgcn_s_wait_asynccnt(0);
#endif
}

// ---------------------------------------------------------------- utilities
__global__ __launch_bounds__(256) void k_zero(float* __restrict__ p, size_t n) {
  size_t i = (size_t)blockIdx.x * 256 + threadIdx.x;
  size_t st = (size_t)gridDim.x * 256;
  for (; i < n; i += st) p[i] = 0.0f;
}

__global__ __launch_bounds__(256) void k_degree(const int* __restrict__ dst,
                                                float* __restrict__ deg, int ne) {
  int e = blockIdx.x * 256 + threadIdx.x;
  if (e < ne) atomicAdd(&deg[dst[e]], 1.0f);
}

__global__ __launch_bounds__(256) void k_invdeg(const float* __restrict__ deg,
                                                float* __restrict__ inv, int n) {
  int i = blockIdx.x * 256 + threadIdx.x;
  if (i < n) inv[i] = 1.0f / fmaxf(deg[i], 1.0f);
}

// ------------------------------------------------- edge gather + scatter-add
// One wave per edge: 2 coalesced 128B loads of h[src], 64 f32 atomics to agg[dst].
__global__ __launch_bounds__(256) void k_scatter(const float* __restrict__ h,
                                                 const int* __restrict__ src,
                                                 const int* __restrict__ dst,
                                                 float* __restrict__ agg, int ne) {
  int e = (blockIdx.x * 256 + threadIdx.x) >> 5;
  int lane = threadIdx.x & 31;
  if (e >= ne) return;
  __builtin_prefetch(src + e + 4096, 0, 0);   // global_prefetch_b8 (speculative)
  __builtin_prefetch(dst + e + 4096, 0, 0);
  size_t s = (size_t)src[e] * HID;
  size_t d = (size_t)dst[e] * HID;
  float v0 = h[s + lane];
  float v1 = h[s + 32 + lane];
  atomicAdd(&agg[d + lane], v0);
  atomicAdd(&agg[d + 32 + lane], v1);
}

// ----------------------------------------------- input projection (K = 128)
// Block = 224 thr = 7 waves, 112 rows; each wave owns one 16-row tile, all 64 cols.
__global__ __launch_bounds__(224) void k_proj(const float* __restrict__ x,
                                              const float* __restrict__ fc_w,
                                              const float* __restrict__ fc_b,
                                              float* __restrict__ h0, int n_nodes) {
  __shared__ float sX[112 * 132];  // +4 pad: conflict-free A-fragment reads
  const int tid = threadIdx.x;
  const int base = blockIdx.x * 112;
  int rows = n_nodes - base;
  if (rows > 112) rows = 112;
  const float4* x4 = (const float4*)x;
  for (int idx = tid; idx < rows * 32; idx += 224) {
    int r = idx >> 5, c4 = idx & 31;
    stage_b128(&x4[(size_t)(base + r) * 32 + c4], &sX[r * 132 + c4 * 4]);
  }
  stage_wait();
  __syncthreads();
  const int wv = tid >> 5, lane = tid & 31;
  const int tile = wv * 16;
  if (base + tile >= n_nodes) return;          // wave-uniform; EXEC stays full for WMMA
  const int hi = lane >> 4, lr = lane & 15;
  v8f acc[4] = {};
  const float* rowX = &sX[(tile + lr) * 132];
  for (int kk = 0; kk < IN_DIM; kk += 4) {
    const int kb = kk + hi * 2;
    v2f a; a.x = rowX[kb]; a.y = rowX[kb + 1];
#pragma unroll
    for (int ct = 0; ct < 4; ++ct) {
      const int n = ct * 16 + lr;              // B[k][n] = fc_w[n][k], cache-resident
      v2f b; b.x = fc_w[n * IN_DIM + kb]; b.y = fc_w[n * IN_DIM + kb + 1];
      acc[ct] = wmma_f32(a, b, acc[ct]);
    }
  }
#pragma unroll
  for (int ct = 0; ct < 4; ++ct) {
    const int col = ct * 16 + lr;
    const float bv = fc_b[col];
#pragma unroll
    for (int r = 0; r < 8; ++r) {
      size_t node = (size_t)(base + tile + r + hi * 8);
      h0[node * HID + col] = fmaxf(acc[ct][r] + bv, 0.0f);
    }
  }
}

// ------------------- SAGE combine: out = (agg*invdeg)@wl^T + h@wr^T + b [, LN+ReLU]
// Row-scaling by invdeg commutes through the GEMM -> applied to A-fragments.
template <int LN>
__global__ __launch_bounds__(224) void k_combine(
    const float* __restrict__ agg, const float* __restrict__ hin,
    const float* __restrict__ invdeg, const float* __restrict__ wl,
    const float* __restrict__ wr, const float* __restrict__ bias,
    const float* __restrict__ ln_g, const float* __restrict__ ln_b,
    float* __restrict__ out, int n_nodes) {
  __shared__ float sA[112 * 68];
  __shared__ float sH[112 * 68];
  const int tid = threadIdx.x;
  const int base = blockIdx.x * 112;
  int rows = n_nodes - base;
  if (rows > 112) rows = 112;
  const float4* a4 = (const float4*)agg;
  const float4* h4 = (const float4*)hin;
  for (int idx = tid; idx < rows * 16; idx += 224) {
    int r = idx >> 4, c4 = idx & 15;
    size_t g = (size_t)(base + r) * 16 + c4;
    stage_b128(&a4[g], &sA[r * 68 + c4 * 4]);
    stage_b128(&h4[g], &sH[r * 68 + c4 * 4]);
  }
  stage_wait();
  __syncthreads();
  const int wv = tid >> 5, lane = tid & 31;
  const int tile = wv * 16;
  if (base + tile >= n_nodes) return;
  const int hi = lane >> 4, lr = lane & 15;
  const float idv = invdeg[base + tile + lr];  // row scale for this lane's A row
  v8f acc[4] = {};
  const float* rowA = &sA[(tile + lr) * 68];
  const float* rowH = &sH[(tile + lr) * 68];
  for (int kk = 0; kk < HID; kk += 4) {
    const int kb = kk + hi * 2;
    v2f aA; aA.x = rowA[kb] * idv; aA.y = rowA[kb + 1] * idv;
    v2f aH; aH.x = rowH[kb]; aH.y = rowH[kb + 1];
#pragma unroll
    for (int ct = 0; ct < 4; ++ct) {
      const int n = ct * 16 + lr;
      v2f bl; bl.x = wl[n * HID + kb]; bl.y = wl[n * HID + kb + 1];
      acc[ct] = wmma_f32(aA, bl, acc[ct]);
      v2f br; br.x = wr[n * HID + kb]; br.y = wr[n * HID + kb + 1];
      acc[ct] = wmma_f32(aH, br, acc[ct]);
    }
  }
  float bb[4], gg[4], be[4];
#pragma unroll
  for (int ct = 0; ct < 4; ++ct) {
    bb[ct] = bias[ct * 16 + lr];
    if (LN) { gg[ct] = ln_g[ct * 16 + lr]; be[ct] = ln_b[ct * 16 + lr]; }
  }
#pragma unroll
  for (int r = 0; r < 8; ++r) {
    size_t node = (size_t)(base + tile + r + hi * 8);
    float v[4];
#pragma unroll
    for (int ct = 0; ct < 4; ++ct) v[ct] = acc[ct][r] + bb[ct];
    if (LN) {
      // Row of 64 cols lives across 4 ct-fragments x 16 lanes of this half-wave.
      float s = v[0] + v[1] + v[2] + v[3];
      s += __shfl_xor(s, 1, 32); s += __shfl_xor(s, 2, 32);
      s += __shfl_xor(s, 4, 32); s += __shfl_xor(s, 8, 32);
      float mu = s * (1.0f / 64.0f);
      float q = 0.0f;
#pragma unroll
      for (int ct = 0; ct < 4; ++ct) { float d0 = v[ct] - mu; q += d0 * d0; }
      q += __shfl_xor(q, 1, 32); q += __shfl_xor(q, 2, 32);
      q += __shfl_xor(q, 4, 32); q += __shfl_xor(q, 8, 32);
      float rn = rsqrtf(q * (1.0f / 64.0f) + 1e-5f);
#pragma unroll
      for (int ct = 0; ct < 4; ++ct) {
        float o = (v[ct] - mu) * rn * gg[ct] + be[ct];
        out[node * HID + ct * 16 + lr] = fmaxf(o, 0.0f);
      }
    } else {
#pragma unroll
      for (int ct = 0; ct < 4; ++ct) out[node * HID + ct * 16 + lr] = v[ct];
    }
  }
}

// --------------------------------------------- graph pooling (batch sorted)
#define POOL_CHUNK 256
__global__ __launch_bounds__(64) void k_pool(const float* __restrict__ ne,
                                             const int* __restrict__ batch,
                                             float* __restrict__ ge, int n_nodes) {
  int f = threadIdx.x;
  int start = blockIdx.x * POOL_CHUNK;
  if (start >= n_nodes) return;
  int end = start + POOL_CHUNK;
  if (end > n_nodes) end = n_nodes;
  float accv = 0.0f;
  int cur = batch[start];
  for (int i = start; i < end; ++i) {
    int b = batch[i];
    if (b != cur) { atomicAdd(&ge[(size_t)cur * HID + f], accv); accv = 0.0f; cur = b; }
    accv += ne[(size_t)i * HID + f];
  }
  atomicAdd(&ge[(size_t)cur * HID + f], accv);
}

// ---------------------------------------------------------------- launcher
extern "C" void kernel_launch(void* const* d_in, const int* in_sizes, int n_in,
                              void* d_out, int out_size, void* d_ws, size_t ws_size,
                              hipStream_t stream) {
  const float* x     = (const float*)d_in[0];
  const int*   ei    = (const int*)d_in[1];
  const int*   batch = (const int*)d_in[2];
  const float* fc_w  = (const float*)d_in[3];
  const float* fc_b  = (const float*)d_in[4];
  const float* wl[3] = {(const float*)d_in[5], (const float*)d_in[8],  (const float*)d_in[11]};
  const float* wr[3] = {(const float*)d_in[6], (const float*)d_in[9],  (const float*)d_in[12]};
  const float* bs[3] = {(const float*)d_in[7], (const float*)d_in[10], (const float*)d_in[13]};
  const float* lng[2] = {(const float*)d_in[14], (const float*)d_in[16]};
  const float* lnb[2] = {(const float*)d_in[15], (const float*)d_in[17]};
  const int* src = ei;
  const int* dst = ei + N_EDGES;

  const size_t NF = (size_t)N_NODES * HID;
  float* ws     = (float*)d_ws;
  float* hA     = ws;            // 6.4M f32
  float* hB     = ws + NF;       // 6.4M f32
  float* agg    = ws + 2 * NF;   // 6.4M f32
  float* deg    = ws + 3 * NF;   // 100K f32
  float* invdeg = deg + N_NODES; // 100K f32

  float* node_out  = (float*)d_out;      // 100000 x 64
  float* graph_out = node_out + NF;      // 64 x 64

  const int nodeBlocks = (N_NODES + 111) / 112;
  const int edgeBlocks = (N_EDGES + 7) / 8;

  // degree / inverse-degree (edge structure only, computed once)
  k_zero<<<512, 256, 0, stream>>>(deg, (size_t)N_NODES);
  k_degree<<<(N_EDGES + 255) / 256, 256, 0, stream>>>(dst, deg, N_EDGES);
  k_invdeg<<<(N_NODES + 255) / 256, 256, 0, stream>>>(deg, invdeg, N_NODES);

  // h = relu(x @ fc_w^T + fc_b)
  k_proj<<<nodeBlocks, 224, 0, stream>>>(x, fc_w, fc_b, hA, N_NODES);

  // layer 0: hA -> hB (LN + ReLU)
  k_zero<<<4096, 256, 0, stream>>>(agg, NF);
  k_scatter<<<edgeBlocks, 256, 0, stream>>>(hA, src, dst, agg, N_EDGES);
  k_combine<1><<<nodeBlocks, 224, 0, stream>>>(agg, hA, invdeg, wl[0], wr[0], bs[0],
                                               lng[0], lnb[0], hB, N_NODES);
  // layer 1: hB -> hA (LN + ReLU)
  k_zero<<<4096, 256, 0, stream>>>(agg, NF);
  k_scatter<<<edgeBlocks, 256, 0, stream>>>(hB, src, dst, agg, N_EDGES);
  k_combine<1><<<nodeBlocks, 224, 0, stream>>>(agg, hB, invdeg, wl[1], wr[1], bs[1],
                                               lng[1], lnb[1], hA, N_NODES);
  // layer 2: hA -> node_embed (no LN/ReLU)
  k_zero<<<4096, 256, 0, stream>>>(agg, NF);
  k_scatter<<<edgeBlocks, 256, 0, stream>>>(hA, src, dst, agg, N_EDGES);
  k_combine<0><<<nodeBlocks, 224, 0, stream>>>(agg, hA, invdeg, wl[2], wr[2], bs[2],
                                               nullptr, nullptr, node_out, N_NODES);

  // graph pooling ('add')
  k_zero<<<16, 256, 0, stream>>>(graph_out, (size_t)N_GRAPHS * HID);
  k_pool<<<(N_NODES + POOL_CHUNK - 1) / POOL_CHUNK, 64, 0, stream>>>(node_out, batch,
                                                                     graph_out, N_NODES);
}